// WaveletTransform_32306744001001
// MI455X (gfx1250) — compile-verified
//
#include <hip/hip_runtime.h>

// Wavelet-style 2x2 stride-2 stencil: out = (3a - b - c - d) * 0.5 over
// clipped 2x2 patches of x (shape 64x3x512x512 -> 64x3x256x256, fp32).
//
// Memory-bound: 251.6 MB moved, ~75 MFLOP => ~10.8 us floor at 23.3 TB/s.
// Strategy: pure streaming kernel. Per thread: two b128 loads from row 2h,
// two from row 2h+1 (even/odd lanes give a,b / c,d), one b128 store of 4
// outputs. Non-temporal TH hints on all traffic (no reuse, stream > L2),
// plus a gfx1250 global_prefetch one row-pair ahead.

typedef float v4f __attribute__((ext_vector_type(4)));

#define IN_W      512                     // input width (floats)
#define IN_PLANE  (512 * 512)             // floats per input plane
#define OUT_W     256
#define OUT_PLANE (256 * 256)             // floats per output plane

__device__ __forceinline__ float clamp01(float v) {
    // lowers to v_med3_f32 / min+max pair
    return __builtin_fminf(__builtin_fmaxf(v, 0.0f), 1.0f);
}

__global__ __launch_bounds__(256) void
WaveletTransform_32306744001001_kernel(const float* __restrict__ x,
                                       float* __restrict__ out,
                                       int n_threads) {
    const int g = blockIdx.x * blockDim.x + threadIdx.x;
    if (g >= n_threads) return;

    // Each thread produces 4 consecutive outputs of one output row.
    // 256 out cols / 4 = 64 groups per row; 256 out rows per plane.
    const unsigned w4  = (unsigned)g & 63u;          // output col group
    const unsigned h   = ((unsigned)g >> 6) & 255u;  // output row
    const unsigned img = (unsigned)g >> 14;          // plane index (N*C)

    // Input row pair 2h, 2h+1; columns [8*w4, 8*w4+8).
    const float* row0 = x + (size_t)img * IN_PLANE
                          + ((size_t)h << 10)        // 2h * 512
                          + ((size_t)w4 << 3);       // 8 * w4
    const float* row1 = row0 + IN_W;

    // Warm GL2 for the next output row's loads (gfx1250 global_prefetch_b8).
    __builtin_prefetch(row0 + 2 * IN_W, 0, 0);

    // Non-temporal streaming loads (zero reuse; stream exceeds 192 MB L2).
    v4f p00 = __builtin_nontemporal_load((const v4f*)row0);        // a0 b0 a1 b1
    v4f p01 = __builtin_nontemporal_load((const v4f*)(row0 + 4));  // a2 b2 a3 b3
    v4f p10 = __builtin_nontemporal_load((const v4f*)row1);        // c0 d0 c1 d1
    v4f p11 = __builtin_nontemporal_load((const v4f*)(row1 + 4));  // c2 d2 c3 d3

    const float a0 = clamp01(p00.x), b0 = clamp01(p00.y);
    const float a1 = clamp01(p00.z), b1 = clamp01(p00.w);
    const float a2 = clamp01(p01.x), b2 = clamp01(p01.y);
    const float a3 = clamp01(p01.z), b3 = clamp01(p01.w);
    const float c0 = clamp01(p10.x), d0 = clamp01(p10.y);
    const float c1 = clamp01(p10.z), d1 = clamp01(p10.w);
    const float c2 = clamp01(p11.x), d2 = clamp01(p11.y);
    const float c3 = clamp01(p11.z), d3 = clamp01(p11.w);

    v4f r;
    r.x = (3.0f * a0 - b0 - c0 - d0) * 0.5f;
    r.y = (3.0f * a1 - b1 - c1 - d1) * 0.5f;
    r.z = (3.0f * a2 - b2 - c2 - d2) * 0.5f;
    r.w = (3.0f * a3 - b3 - c3 - d3) * 0.5f;

    float* o = out + (size_t)img * OUT_PLANE
                   + ((size_t)h << 8)                // h * 256
                   + ((size_t)w4 << 2);              // 4 * w4
    __builtin_nontemporal_store(r, (v4f*)o);
}

extern "C" void kernel_launch(void* const* d_in, const int* in_sizes, int n_in,
                              void* d_out, int out_size, void* d_ws, size_t ws_size,
                              hipStream_t stream) {
    const float* x = (const float*)d_in[0];
    float* out     = (float*)d_out;

    // 4 outputs per thread; out_size = 64*3*256*256 = 12,582,912
    const int n_threads = out_size / 4;            // 3,145,728
    const int block = 256;                         // 8 wave32 waves
    const int grid  = (n_threads + block - 1) / block;

    WaveletTransform_32306744001001_kernel<<<grid, block, 0, stream>>>(x, out, n_threads);
}